// OptLayer_34763465293856
// MI455X (gfx1250) — compile-verified
//
#include <hip/hip_runtime.h>
#include <stdint.h>

#define DD   6
#define SDIM 17
#define CAPV 20.0f
#define BIGV 1e30f

// Low 32 bits of a generic shared-memory address == LDS byte offset
// (gfx1250 flat aperture lives in addr[63:32]).
__device__ __forceinline__ uint32_t lds_off(const void* p) {
  return (uint32_t)(uintptr_t)p;
}

// CDNA5 async global->LDS copies, GVS addressing form:
//   global addr = SGPR_base + voffset + INST_OFFSET
//   LDS addr: vdst VGPR value plus the same offset   (per ISA pseudocode the
//                instruction offset applies to BOTH addresses)
// Tracked by ASYNCcnt.
#define ASYNC_B64_GVS(ldsv, voff, sbase, OFFLIT)                      \
  asm volatile("global_load_async_to_lds_b64 %0, %1, %2 offset:" OFFLIT \
               :: "v"(ldsv), "v"(voff), "s"(sbase) : "memory")
#define ASYNC_B32_GVS(ldsv, voff, sbase, OFFLIT)                      \
  asm volatile("global_load_async_to_lds_b32 %0, %1, %2 offset:" OFFLIT \
               :: "v"(ldsv), "v"(voff), "s"(sbase) : "memory")

__device__ __forceinline__ void wait_async0() {
  asm volatile("s_wait_asynccnt 0x0" ::: "memory");
}

__global__ __launch_bounds__(256) void proj_wl1_kernel(
    const float* __restrict__ x, const float* __restrict__ s,
    float* __restrict__ out, int n) {
  __shared__ float xs[256 * DD];  // 6 KB
  __shared__ float ss[256 * DD];  // 6 KB
  const int tid = threadIdx.x;
  const int row = blockIdx.x * 256 + tid;

  if (row < n) {
    const uint32_t xoff = (uint32_t)row * 24u;  // byte offset of x row
    const uint32_t soff = (uint32_t)row * 68u;  // byte offset of s row
    const uint32_t lx = lds_off(&xs[tid * DD]);        // 8B aligned (tid*24)
    const uint32_t lw = lds_off(&ss[tid * DD]) - 44u;  // pre-biased for offset:44..64

    // x row: 24B as 3x async B64; INST_OFFSET advances global+LDS together.
    ASYNC_B64_GVS(lx, xoff, x, "0");
    ASYNC_B64_GVS(lx, xoff, x, "8");
    ASYNC_B64_GVS(lx, xoff, x, "16");
    // s[row, 11..16]: 6x async B32 gather (only 4B aligned).
    ASYNC_B32_GVS(lw, soff, s, "44");
    ASYNC_B32_GVS(lw, soff, s, "48");
    ASYNC_B32_GVS(lw, soff, s, "52");
    ASYNC_B32_GVS(lw, soff, s, "56");
    ASYNC_B32_GVS(lw, soff, s, "60");
    ASYNC_B32_GVS(lw, soff, s, "64");
  }
  // Wait on this wave's own async copies; each thread only reads the LDS
  // bytes its own lanes wrote, so no cross-wave barrier is required.
  wait_async0();
  if (row >= n) return;

  float q[DD], w[DD], a[DD], r[DD];
#pragma unroll
  for (int j = 0; j < DD; ++j) {
    q[j] = xs[tid * DD + j];
    w[j] = fabsf(ss[tid * DD + j]);
    a[j] = fabsf(q[j]);
  }

  float total = 0.0f;
#pragma unroll
  for (int j = 0; j < DD; ++j) total += w[j] * a[j];

#pragma unroll
  for (int j = 0; j < DD; ++j)
    r[j] = (w[j] > 0.0f) ? (a[j] / w[j]) : BIGV;

  // Descending sort by r, carrying (w, a). Bose-Nelson n=6, 12 comparators.
#define CE(i, jj)                                              \
  {                                                            \
    if (r[i] < r[jj]) {                                        \
      float t;                                                 \
      t = r[i]; r[i] = r[jj]; r[jj] = t;                       \
      t = w[i]; w[i] = w[jj]; w[jj] = t;                       \
      t = a[i]; a[i] = a[jj]; a[jj] = t;                       \
    }                                                          \
  }
  CE(1, 2) CE(4, 5)
  CE(0, 2) CE(3, 5)
  CE(0, 1) CE(3, 4) CE(2, 5)
  CE(0, 3) CE(1, 4)
  CE(2, 4) CE(1, 3)
  CE(2, 3)
#undef CE

  // Prefix sums -> candidate lambdas; count valid prefixes (matches reference).
  float cwq = 0.0f, cw2 = 0.0f;
  float lamk[DD];
  int cnt = 0;
#pragma unroll
  for (int k = 0; k < DD; ++k) {
    cwq += w[k] * a[k];
    cw2 += w[k] * w[k];
    const float cs = (cw2 > 0.0f) ? cw2 : 1.0f;
    const float lk = (cwq - CAPV) / cs;
    lamk[k] = lk;
    cnt += ((r[k] > lk) && (cw2 > 0.0f)) ? 1 : 0;
  }

  float lam = lamk[0];
#pragma unroll
  for (int k = 1; k < DD; ++k) lam = (cnt > k) ? lamk[k] : lam;
  lam = (total > CAPV) ? fmaxf(lam, 0.0f) : 0.0f;

  float o[DD];
#pragma unroll
  for (int j = 0; j < DD; ++j)
    o[j] = copysignf(fmaxf(a[j] - lam * w[j], 0.0f), q[j]);

  // 24B/row output, 8B-aligned -> three float2 stores.
  float2* op = (float2*)(out + (size_t)row * DD);
  op[0] = make_float2(o[0], o[1]);
  op[1] = make_float2(o[2], o[3]);
  op[2] = make_float2(o[4], o[5]);
}

extern "C" void kernel_launch(void* const* d_in, const int* in_sizes, int n_in,
                              void* d_out, int out_size, void* d_ws, size_t ws_size,
                              hipStream_t stream) {
  const float* x = (const float*)d_in[0];
  const float* s = (const float*)d_in[1];
  float* out = (float*)d_out;
  const int n = in_sizes[0] / DD;  // number of rows (B)
  const int blocks = (n + 255) / 256;
  hipLaunchKernelGGL(proj_wl1_kernel, dim3(blocks), dim3(256), 0, stream,
                     x, s, out, n);
}